// LaplacianLoss_63806034149711
// MI455X (gfx1250) — compile-verified
//
#include <hip/hip_runtime.h>

typedef float v2f __attribute__((ext_vector_type(2)));
typedef float v8f __attribute__((ext_vector_type(8)));

#define SCAN_T 1024
#define LDS_CAP 4096   // max CSR entries staged per block (32 KB LDS)

// ---------- CSR construction ----------

__global__ void k_zero(int* __restrict__ p, int n) {
  int g = blockIdx.x * blockDim.x + threadIdx.x;
  if (g < n) p[g] = 0;
}

__global__ void k_hist(const int* __restrict__ rows, int* __restrict__ counts, int n) {
  int g = blockIdx.x * blockDim.x + threadIdx.x;
  if (g < n) atomicAdd(&counts[rows[g]], 1);
}

// Per-block exclusive scan of 1024 elements; emits block totals.
__global__ void k_scan_block(const int* __restrict__ in, int* __restrict__ out,
                             int* __restrict__ sums, int n) {
  __shared__ int tmp[SCAN_T];
  int t = threadIdx.x;
  int g = blockIdx.x * SCAN_T + t;
  int v = (g < n) ? in[g] : 0;
  tmp[t] = v;
  __syncthreads();
  for (int off = 1; off < SCAN_T; off <<= 1) {
    int x = (t >= off) ? tmp[t - off] : 0;
    __syncthreads();
    tmp[t] += x;
    __syncthreads();
  }
  if (g < n) out[g] = tmp[t] - v;          // exclusive
  if (t == SCAN_T - 1) sums[blockIdx.x] = tmp[t];
}

// Single-block exclusive scan of the block totals (nb <= 1024).
__global__ void k_scan_sums(int* __restrict__ sums, int nb) {
  __shared__ int tmp[SCAN_T];
  int t = threadIdx.x;
  int v = (t < nb) ? sums[t] : 0;
  tmp[t] = v;
  __syncthreads();
  for (int off = 1; off < SCAN_T; off <<= 1) {
    int x = (t >= off) ? tmp[t - off] : 0;
    __syncthreads();
    tmp[t] += x;
    __syncthreads();
  }
  if (t < nb) sums[t] = tmp[t] - v;
}

__global__ void k_add_off(int* __restrict__ rp, const int* __restrict__ sums, int V, int nE) {
  int g = blockIdx.x * blockDim.x + threadIdx.x;
  if (g < V) rp[g] += sums[g / SCAN_T];
  if (g == 0) rp[V] = nE;
}

__global__ void k_scatter(const int* __restrict__ rows, const int* __restrict__ cols,
                          const float* __restrict__ vals, const int* __restrict__ rp,
                          int* __restrict__ cursor, int* __restrict__ colsS,
                          float* __restrict__ valsS, int n) {
  int g = blockIdx.x * blockDim.x + threadIdx.x;
  if (g < n) {
    int r = rows[g];
    int p = rp[r] + atomicAdd(&cursor[r], 1);
    colsS[p] = cols[g];
    valsS[p] = vals[g];
  }
}

// ---------- CDNA5 async global->LDS copy (ASYNCcnt path) ----------

// Per-lane 4B copy from global memory at sbase plus byte offset voff to the LDS address in lds_off
__device__ __forceinline__ void async_g2l_b32(unsigned lds_off, const void* sbase, int voff) {
  asm volatile("global_load_async_to_lds_b32 %0, %1, %2"
               :: "v"(lds_off), "v"(voff), "s"(sbase)
               : "memory");
}

__device__ __forceinline__ void wait_async0() {
  asm volatile("s_wait_asynccnt 0" ::: "memory");
}

// ---------- main pass ----------

// Wave32 sum via V_WMMA_F32_16X16X4_F32.
// A (16x4) layout: lanes 0-15 {V0=K0,V1=K1}, lanes 16-31 {V0=K2,V1=K3}.
// A = (x, 0) per lane, B = ones  =>  D[m,n] = x[m] + x[m+16] for every n.
// Summing the 8 D registers gives the half-wave sum per lane; one xor-16
// shuffle completes the full wave sum. EXEC must be all ones here.
__device__ __forceinline__ float wave_sum_wmma(float x) {
  v2f a; a.x = x;    a.y = 0.0f;
  v2f b; b.x = 1.0f; b.y = 1.0f;
  v8f c = {0.f, 0.f, 0.f, 0.f, 0.f, 0.f, 0.f, 0.f};
  v8f d = __builtin_amdgcn_wmma_f32_16x16x4_f32(false, a, false, b, (short)0, c, false, false);
  float h = d[0] + d[1] + d[2] + d[3] + d[4] + d[5] + d[6] + d[7];
  h += __shfl_xor(h, 16, 32);
  return h;
}

__global__ void k_main(const float* __restrict__ verts, const int* __restrict__ rp,
                       const int* __restrict__ colsS, const float* __restrict__ valsS,
                       float* __restrict__ partials, int V) {
  __shared__ int   sh_cols[LDS_CAP];
  __shared__ float sh_vals[LDS_CAP];
  __shared__ float lsum[32];

  int tid = threadIdx.x;
  int v0  = blockIdx.x * blockDim.x;
  int v   = v0 + tid;
  int bb  = blockIdx.y;
  const float* vb = verts + (size_t)bb * (size_t)V * 3u;

  // Block's contiguous CSR span [s0, s1)
  int vLast = v0 + (int)blockDim.x; if (vLast > V) vLast = V;
  int s0 = rp[v0];
  int s1 = rp[vLast];
  int len = s1 - s0;
  bool staged = (len <= LDS_CAP);          // block-uniform branch

  if (staged) {
    const int* gcols = colsS + s0;
    const float* gvals = valsS + s0;
    for (int i = tid; i < len; i += (int)blockDim.x) {
      async_g2l_b32((unsigned)(size_t)&sh_cols[i], gcols, i * 4);
      async_g2l_b32((unsigned)(size_t)&sh_vals[i], gvals, i * 4);
    }
    wait_async0();                         // this wave's async copies done
  }
  __syncthreads();                         // LDS visible block-wide

  float ax = 0.f, ay = 0.f, az = 0.f;
  if (v < V) {
    size_t p = (size_t)v * 3u;
    ax = vb[p]; ay = vb[p + 1]; az = vb[p + 2];   // identity diagonal
    int s = rp[v], e = rp[v + 1];
    if (staged) {
      for (int k = s; k < e; ++k) {
        int cc  = sh_cols[k - s0];
        float w = sh_vals[k - s0];
        size_t q = (size_t)cc * 3u;
        ax = fmaf(w, vb[q],     ax);
        ay = fmaf(w, vb[q + 1], ay);
        az = fmaf(w, vb[q + 2], az);
      }
    } else {
      for (int k = s; k < e; ++k) {
        int cc  = colsS[k];
        float w = valsS[k];
        size_t q = (size_t)cc * 3u;
        ax = fmaf(w, vb[q],     ax);
        ay = fmaf(w, vb[q + 1], ay);
        az = fmaf(w, vb[q + 2], az);
      }
    }
  }
  float sq = ax * ax + ay * ay + az * az;         // 0 for padded lanes

  float wsum = wave_sum_wmma(sq);                 // all lanes converged here

  int lane = tid & 31;
  int wid  = tid >> 5;
  if (lane == 0) lsum[wid] = wsum;
  __syncthreads();
  if (tid == 0) {
    float tot = 0.f;
    int nw = (int)blockDim.x >> 5;
    for (int i = 0; i < nw; ++i) tot += lsum[i];
    partials[(size_t)blockIdx.y * gridDim.x + blockIdx.x] = tot;
  }
}

// Deterministic fixed-order final reduction; fully overwrites d_out.
__global__ void k_finalize(const float* __restrict__ partials, int n, float inv,
                           float* __restrict__ out) {
  __shared__ float tmp[256];
  int t = threadIdx.x;
  float s = 0.f;
  for (int i = t; i < n; i += 256) s += partials[i];
  tmp[t] = s;
  __syncthreads();
  for (int off = 128; off > 0; off >>= 1) {
    if (t < off) tmp[t] += tmp[t + off];
    __syncthreads();
  }
  if (t == 0) out[0] = tmp[0] * inv;
}

// ---------- launch ----------

extern "C" void kernel_launch(void* const* d_in, const int* in_sizes, int n_in,
                              void* d_out, int out_size, void* d_ws, size_t ws_size,
                              hipStream_t stream) {
  (void)n_in; (void)out_size; (void)ws_size;

  const float* verts = (const float*)d_in[0];   // (B, V, 3) f32
  const int*   rows  = (const int*)d_in[1];     // (2E,) i32
  const int*   cols  = (const int*)d_in[2];     // (2E,) i32
  const float* vals  = (const float*)d_in[3];   // (2E,) f32
  float* out = (float*)d_out;

  const int Bm = 32;                            // batch size per reference
  int V  = in_sizes[0] / (Bm * 3);
  int nE = in_sizes[1];

  // Workspace layout (~15 MB): counts/cursor | row_ptr | scan sums | colsS | valsS | partials
  int* counts   = (int*)d_ws;
  int* rp       = counts + V;
  int* sums     = rp + V + 1;
  int* colsS    = sums + SCAN_T;
  float* valsS  = (float*)(colsS + nE);
  float* parts  = valsS + nE;

  int nScan   = (V + SCAN_T - 1) / SCAN_T;      // 256 for V = 262144
  int gx      = (V + 255) / 256;
  int nBlocks = gx * Bm;

  k_zero      <<<(V  + 255) / 256, 256, 0, stream>>>(counts, V);
  k_hist      <<<(nE + 255) / 256, 256, 0, stream>>>(rows, counts, nE);
  k_scan_block<<<nScan, SCAN_T, 0, stream>>>(counts, rp, sums, V);
  k_scan_sums <<<1, SCAN_T, 0, stream>>>(sums, nScan);
  k_add_off   <<<(V  + 255) / 256, 256, 0, stream>>>(rp, sums, V, nE);
  k_zero      <<<(V  + 255) / 256, 256, 0, stream>>>(counts, V);   // reset cursors
  k_scatter   <<<(nE + 255) / 256, 256, 0, stream>>>(rows, cols, vals, rp, counts,
                                                     colsS, valsS, nE);
  k_main      <<<dim3(gx, Bm), 256, 0, stream>>>(verts, rp, colsS, valsS, parts, V);

  float inv = 1.0f / ((float)Bm * (float)V);
  k_finalize  <<<1, 256, 0, stream>>>(parts, nBlocks, inv, out);
}